// Hyperbolic_Lines_1803886265743
// MI455X (gfx1250) — compile-verified
//
#include <hip/hip_runtime.h>
#include <cstdint>

// Reference: y [N, D] fp32, w [D] fp32; loss = sum(acosh(1 + d2)^2).
#define N_ROWS 500000
#define DIMS   128

constexpr int WAVE       = 32;                     // single wave32 per block
constexpr int TILE_ROWS  = 16;                     // rows per DMA tile (divides N)
constexpr int ROW_BYTES  = DIMS * 4;               // 512
constexpr int TILE_BYTES = TILE_ROWS * ROW_BYTES;  // 8192
constexpr int PITCH_F    = 132;                    // padded floats/row in LDS
constexpr int PITCH_B    = PITCH_F * 4;            // 528 (bank-conflict floor)
constexpr int PIPE       = 3;                      // triple-buffered DMA
constexpr int GRID       = 4096;
static_assert(PIPE * TILE_ROWS <= 63, "ASYNCcnt is 6 bits");

__device__ __forceinline__ float wave_sum32(float v) {
#pragma unroll
  for (int m = 16; m > 0; m >>= 1) v += __shfl_xor(v, m, 32);
  return v;  // all lanes hold the total
}

__global__ __launch_bounds__(WAVE)
void hyper_lines_main(const float* __restrict__ w,
                      const float* __restrict__ y,
                      float* __restrict__ partial,
                      int nTiles)
{
  // 3 x 16 rows x 132 floats = ~25 KB: ~12 blocks/WGP of the 320 KB LDS.
  __shared__ __align__(16) float tile[PIPE][TILE_ROWS * PITCH_F];

  const int lane = (int)threadIdx.x;
  const int half = lane >> 4;   // 0: cols 0..63, 1: cols 64..127
  const int r    = lane & 15;   // row within tile

  // This lane's 64-float slice of w, held in registers (16 x float4).
  const float4* w4 = reinterpret_cast<const float4*>(w);
  float4 wreg[16];
#pragma unroll
  for (int k = 0; k < 16; ++k) wreg[k] = w4[half * 16 + k];

  float pw = 0.0f;
#pragma unroll
  for (int k = 0; k < 16; ++k) {
    pw = fmaf(wreg[k].x, wreg[k].x, pw);
    pw = fmaf(wreg[k].y, wreg[k].y, pw);
    pw = fmaf(wreg[k].z, wreg[k].z, pw);
    pw = fmaf(wreg[k].w, wreg[k].w, pw);
  }
  pw += __shfl_xor(pw, 16, 32);        // combine halves -> ||w||^2 in all lanes
  const float inv_nw2 = 1.0f / pw;

  // Low 32 bits of a generic LDS pointer == wave-relative LDS byte address.
  uint32_t ldsBase[PIPE];
#pragma unroll
  for (int b = 0; b < PIPE; ++b)
    ldsBase[b] = (uint32_t)(uintptr_t)(const void*)&tile[b][0];

  const uint32_t limitOff = (uint32_t)N_ROWS * (uint32_t)ROW_BYTES - 16u;

  // One async b128 per lane per row: the 32 lanes cover one 512 B row
  // contiguously in global memory, scattered to the padded LDS row.
  auto issue_tile = [&](int t, int buf) {
    const uint32_t gbase = (uint32_t)t * (uint32_t)TILE_BYTES + (uint32_t)lane * 16u;
    const uint32_t lbase = ldsBase[buf] + (uint32_t)lane * 16u;
#pragma unroll
    for (int j = 0; j < TILE_ROWS; ++j) {
      uint32_t goff = gbase + (uint32_t)j * (uint32_t)ROW_BYTES;
      if (goff > limitOff) goff = limitOff;  // dead for N=500000, keeps counts uniform
      const uint32_t laddr = lbase + (uint32_t)j * (uint32_t)PITCH_B;
      asm volatile("global_load_async_to_lds_b128 %0, %1, %2"
                   :: "v"(laddr), "v"(goff), "s"(y) : "memory");
    }
  };

  const int base   = (int)blockIdx.x;
  const int stride = (int)gridDim.x;
  const int cnt    = (base < nTiles) ? ((nTiles - 1 - base) / stride + 1) : 0;

  const int prime = cnt < PIPE ? cnt : PIPE;
  for (int i = 0; i < prime; ++i) issue_tile(base + i * stride, i);

  float acc = 0.0f;
  for (int i = 0; i < cnt; ++i) {
    // Tiles issued beyond the current one (uniform scalar): pick the wait.
    const int ahead = (cnt - 1 - i) < (PIPE - 1) ? (cnt - 1 - i) : (PIPE - 1);
    if (ahead >= 2)      asm volatile("s_wait_asynccnt 32" ::: "memory");
    else if (ahead == 1) asm volatile("s_wait_asynccnt 16" ::: "memory");
    else                 asm volatile("s_wait_asynccnt 0"  ::: "memory");

    const int t = base + i * stride;
    const float* buf = &tile[i % PIPE][0];
    // Lane (r, half): banks 4r+4k -> lanes 0..15 tile all 64 banks once,
    // lanes 16..31 repeat them: the 2-cycle minimum for 512 B, no hot bank.
    const float4* rowp = reinterpret_cast<const float4*>(buf + r * PITCH_F + half * 64);

    // Four independent FMA chains (2 per reduction) so ~3 waves/SIMD can
    // hide the FMA latency despite the low LDS-limited occupancy.
    float syw0 = 0.0f, syw1 = 0.0f, syy0 = 0.0f, syy1 = 0.0f;
#pragma unroll
    for (int k = 0; k < 16; k += 2) {
      const float4 a = rowp[k];
      const float4 b = rowp[k + 1];
      syw0 = fmaf(a.x, wreg[k].x, syw0);
      syw0 = fmaf(a.y, wreg[k].y, syw0);
      syw0 = fmaf(a.z, wreg[k].z, syw0);
      syw0 = fmaf(a.w, wreg[k].w, syw0);
      syy0 = fmaf(a.x, a.x, syy0);
      syy0 = fmaf(a.y, a.y, syy0);
      syy0 = fmaf(a.z, a.z, syy0);
      syy0 = fmaf(a.w, a.w, syy0);
      syw1 = fmaf(b.x, wreg[k + 1].x, syw1);
      syw1 = fmaf(b.y, wreg[k + 1].y, syw1);
      syw1 = fmaf(b.z, wreg[k + 1].z, syw1);
      syw1 = fmaf(b.w, wreg[k + 1].w, syw1);
      syy1 = fmaf(b.x, b.x, syy1);
      syy1 = fmaf(b.y, b.y, syy1);
      syy1 = fmaf(b.z, b.z, syy1);
      syy1 = fmaf(b.w, b.w, syy1);
    }
    float syw = syw0 + syw1;
    float syy = syy0 + syy1;

    // Single cross-lane exchange per 16 rows: combine the two half-rows.
    syw += __shfl_xor(syw, 16, 32);
    syy += __shfl_xor(syy, 16, 32);

    const long grow = (long)t * TILE_ROWS + r;
    if (half == 0 && grow < (long)N_ROWS) {      // 16 rows' acosh^2 in parallel
      const float d2 = fmaxf(syy - syw * syw * inv_nw2, 0.0f);
      const float x  = 1.0f + d2;
      const float a  = logf(x + sqrtf(x * x - 1.0f));   // acosh(x)
      acc = fmaf(a, a, acc);
    }

    const int j = i + PIPE;
    if (j < cnt) {
      // All ds reads of this buffer were consumed above; drain before the
      // DMA-side writes can land in the same buffer.
      asm volatile("s_wait_dscnt 0" ::: "memory");
      issue_tile(base + j * stride, j % PIPE);
    }
  }

  const float s = wave_sum32(acc);   // lanes >=16 contributed 0
  if (lane == 0) partial[blockIdx.x] = s;   // deterministic, no atomics
}

__global__ __launch_bounds__(32)
void hyper_lines_finalize(const float* __restrict__ partial, int n,
                          float* __restrict__ out)
{
  const int lane = (int)threadIdx.x;
  float s = 0.0f;
  for (int j = lane; j < n; j += 32) s += partial[j];  // fixed order
  s = wave_sum32(s);
  if (lane == 0) out[0] = s;
}

extern "C" void kernel_launch(void* const* d_in, const int* in_sizes, int n_in,
                              void* d_out, int out_size, void* d_ws, size_t ws_size,
                              hipStream_t stream)
{
  (void)in_sizes; (void)n_in; (void)out_size; (void)ws_size;
  const float* w = (const float*)d_in[0];   // [128]
  const float* y = (const float*)d_in[1];   // [500000, 128]
  float* partial = (float*)d_ws;            // GRID floats scratch (16 KB)
  const int nTiles = (N_ROWS + TILE_ROWS - 1) / TILE_ROWS;  // 31250

  hyper_lines_main<<<GRID, WAVE, 0, stream>>>(w, y, partial, nTiles);
  hyper_lines_finalize<<<1, 32, 0, stream>>>(partial, GRID, (float*)d_out);
}